// StatDenoiser_22342419874368
// MI455X (gfx1250) — compile-verified
//
#include <hip/hip_runtime.h>
#include <stdint.h>

// ---------------- StatDenoiser for MI455X (gfx1250) ----------------
// 147-tap (7x7x3) statistically-gated bilateral filter, (C,H,W,T)=(.,64,64,16).
// Entire problem is L2-resident; kernel is VALU-bound. Halo tiles are staged
// into LDS with CDNA5 per-lane async global->LDS loads (ASYNCcnt), compute is
// fp32 VALU with compile-time-constant LDS offsets (base + imm16 addressing).

#define TH 8
#define TW 2
#define TT 8
#define HH (TH + 6)        // 14
#define WW (TW + 6)        // 8  (power of two -> hw>>3 / hw&7 staging split)
#define TTH (TT + 2)       // 10
#define NCH 18             // 3 img + 9 guidance + 3 estimands + 3 variance
#define CH_STRIDE (HH * WW * TTH)   // 1120 floats per channel
#define NTHREADS (TH * TW * TT)     // 128 threads, 1 output pixel each

// Per-lane async copy of one f32 from global (SGPR base + VGPR byte offset)
// into LDS (per-lane LDS byte address). Tracked by ASYNCcnt.
__device__ __forceinline__ void async_f32_to_lds(uint32_t lds_byte_addr,
                                                 int g_byte_off,
                                                 const float* base) {
  uint64_t b = (uint64_t)(uintptr_t)base;
  asm volatile("global_load_async_to_lds_b32 %0, %1, %2"
               :: "v"(lds_byte_addr), "v"(g_byte_off), "s"(b)
               : "memory");
}

__global__ __launch_bounds__(NTHREADS)
void statdenoiser_gfx1250_kernel(const float* __restrict__ img,
                                 const float* __restrict__ gui,
                                 const float* __restrict__ est,
                                 const float* __restrict__ var,
                                 float* __restrict__ out) {
  // channel order: [0..2]=image, [3..11]=guidance, [12..14]=estimands, [15..17]=variance
  __shared__ float s[NCH][HH][WW][TTH];   // 80,640 bytes

  const int tid = threadIdx.x;
  const int h0 = blockIdx.y * TH;
  const int w0 = blockIdx.x * TW;
  const int t0 = blockIdx.z * TT;

  // ---------------- Stage halo: async global -> LDS ----------------
  // One (hh,ww) column per thread (HH*WW = 112 <= 128 threads), t unrolled.
  // Global base pointer per channel is wave-uniform ("s" constraint).
  const int hw = tid;                 // hw = hh*WW + ww
  const int hh_s = hw >> 3;           // WW == 8
  const int ww_s = hw & 7;
  const int gh_s = h0 - 3 + hh_s;
  const int gw_s = w0 - 3 + ww_s;
  const bool hwin = (hw < HH * WW) && ((unsigned)gh_s < 64u) && ((unsigned)gw_s < 64u);
  const bool hvalid = (hw < HH * WW);
  const int gcol = (gh_s * 1024 + gw_s * 16 + (t0 - 1)) * 4;  // byte off of tt=0
  const uint32_t lds_base = (uint32_t)(uintptr_t)&s[0][0][0][0];

  for (int c = 0; c < NCH; ++c) {
    const float* bp;
    if (c < 3)       bp = img + c * 65536;
    else if (c < 12) bp = gui + (c - 3) * 65536;
    else if (c < 15) bp = est + (c - 12) * 65536;
    else             bp = var + (c - 15) * 65536;

    const uint32_t lds_col = lds_base + (uint32_t)((c * HH * WW + hw) * TTH) * 4u;
#pragma unroll
    for (int tt = 0; tt < TTH; ++tt) {
      // t-range test is block-uniform (t0 uniform, tt compile-time)
      const bool tin = ((unsigned)(t0 - 1 + tt) < 16u);
      if (hwin && tin) {
        async_f32_to_lds(lds_col + 4u * tt, gcol + 4 * tt, bp);
      } else if (hvalid) {
        ((float*)&s[0][0][0][0])[(c * HH * WW + hw) * TTH + tt] = 0.0f;  // zero-pad
      }
    }
  }
  asm volatile("s_wait_asynccnt 0x0" ::: "memory");  // our wave's async LDS writes landed
  __syncthreads();                                   // all waves' halo visible

  // ---------------- Per-thread output pixel ----------------
  const int tl  = tid & (TT - 1);     // t in tile (fast lanes: coalesced, bank-clean)
  const int pix = tid >> 3;
  const int wl  = pix & (TW - 1);
  const int hl  = pix >> 1;           // 0..7

  // Thread base pointer: all taps are pbase + compile-time-constant offsets.
  const float* pbase = &s[0][hl][wl][tl];
  const int CTR = (3 * WW + 3) * TTH + 1;   // center tap offset within a channel

  float cg[9], ce[3], cv[3];
#pragma unroll
  for (int c = 0; c < 9; ++c) cg[c] = pbase[(3 + c) * CH_STRIDE + CTR];
#pragma unroll
  for (int c = 0; c < 3; ++c) {
    ce[c] = pbase[(12 + c) * CH_STRIDE + CTR];
    cv[c] = pbase[(15 + c) * CH_STRIDE + CTR];
  }

  // K[c] = -0.5 * sigma_inv[c] * log2(e): bilateral folded into a single exp2.
  const float K[9] = {-0.07213475f, -0.07213475f, -0.07213475f,
                      -36.0673760f, -36.0673760f, -36.0673760f,
                      -7.21347520f, -7.21347520f, -7.21347520f};
  const float G2 = 8.450649f;  // GAMMA_W^2
  // Welch-test algebra: t^2 = d^2/(vi+vj); member iff d^2 < G2*(vi+vj),
  // zero-variance override: (vi==0 || vj==0) -> member iff d==0.
  // Center tap gets weight exactly 1 (matches the reference's forcing).

  float wsum = 0.0f;
  float oacc[3] = {0.0f, 0.0f, 0.0f};

  const float* p = pbase;            // advances by WW*TTH floats per dh
  for (int dh = 0; dh < 7; ++dh) {
#pragma unroll
    for (int dw = 0; dw < 7; ++dw) {
      const int o = dw * TTH;

      // Bilateral exponents for the three temporal taps (loads are q[0..2]:
      // 12 contiguous bytes -> mergeable DS loads, base + imm16 addressing).
      float e0 = 0.0f, e1 = 0.0f, e2 = 0.0f;
#pragma unroll
      for (int c = 0; c < 9; ++c) {
        const float* q = p + (3 + c) * CH_STRIDE + o;
        const float d0 = q[0] - cg[c];
        const float d1 = q[1] - cg[c];
        const float d2 = q[2] - cg[c];
        e0 = __builtin_fmaf(K[c], d0 * d0, e0);
        e1 = __builtin_fmaf(K[c], d1 * d1, e1);
        e2 = __builtin_fmaf(K[c], d2 * d2, e2);
      }
      const float b0 = exp2f(e0), b1 = exp2f(e1), b2 = exp2f(e2);

      // Statistical membership gates
      bool m0 = true, m1 = true, m2 = true;
#pragma unroll
      for (int c = 0; c < 3; ++c) {
        const float* qe = p + (12 + c) * CH_STRIDE + o;
        const float* qv = p + (15 + c) * CH_STRIDE + o;
        const float vi = cv[c], cec = ce[c];
        {
          const float d = qe[0] - cec, d2 = d * d, vj = qv[0];
          const bool zc = (vi == 0.0f) || (vj == 0.0f);
          m0 = m0 && (zc ? (d2 == 0.0f) : (d2 < G2 * (vi + vj)));
        }
        {
          const float d = qe[1] - cec, d2 = d * d, vj = qv[1];
          const bool zc = (vi == 0.0f) || (vj == 0.0f);
          m1 = m1 && (zc ? (d2 == 0.0f) : (d2 < G2 * (vi + vj)));
        }
        {
          const float d = qe[2] - cec, d2 = d * d, vj = qv[2];
          const bool zc = (vi == 0.0f) || (vj == 0.0f);
          m2 = m2 && (zc ? (d2 == 0.0f) : (d2 < G2 * (vi + vj)));
        }
      }

      const float w0 = m0 ? b0 : 0.0f;
      const float w1 = m1 ? b1 : 0.0f;
      const float w2 = m2 ? b2 : 0.0f;
      wsum += (w0 + w1 + w2);
#pragma unroll
      for (int c = 0; c < 3; ++c) {
        const float* qi = p + c * CH_STRIDE + o;
        oacc[c] = __builtin_fmaf(w0, qi[0],
                  __builtin_fmaf(w1, qi[1],
                  __builtin_fmaf(w2, qi[2], oacc[c])));
      }
    }
    p += WW * TTH;   // next dh row
  }

  const int gh = h0 + hl, gw = w0 + wl, gt = t0 + tl;
  const int oidx = gh * 1024 + gw * 16 + gt;
  const float inv = 1.0f / wsum;     // wsum >= 1 (center weight is exactly 1)
  out[oidx]             = oacc[0] * inv;
  out[65536 + oidx]     = oacc[1] * inv;
  out[2 * 65536 + oidx] = oacc[2] * inv;
}

extern "C" void kernel_launch(void* const* d_in, const int* in_sizes, int n_in,
                              void* d_out, int out_size, void* d_ws, size_t ws_size,
                              hipStream_t stream) {
  (void)in_sizes; (void)n_in; (void)out_size; (void)d_ws; (void)ws_size;
  const float* img = (const float*)d_in[0];  // (3,64,64,16)
  const float* gui = (const float*)d_in[1];  // (9,64,64,16)
  const float* est = (const float*)d_in[2];  // (3,64,64,16)
  const float* var = (const float*)d_in[3];  // (3,64,64,16)
  float* out = (float*)d_out;                // (3,64,64,16)

  dim3 grid(64 / TW, 64 / TH, 16 / TT);      // (32, 8, 2) = 512 blocks
  statdenoiser_gfx1250_kernel<<<grid, NTHREADS, 0, stream>>>(img, gui, est, var, out);
}